// PositionBias_26637387170463
// MI455X (gfx1250) — compile-verified
//
#include <hip/hip_runtime.h>
#include <hip/hip_bf16.h>
#include <stdint.h>

// out[b,j,m,k] = x[b,j,m,k] + T[min(|sd_b - j + m|, 127), k]
// B=8, Cj=256, Cm=256, K=64 bins, bias table 128x64 f32 (32 KB).

#define BIAS_ROWS   128
#define NUM_BINS    64
#define DIM         256            // Cj == Cm == 256
#define ROWS_PER_BLOCK 4
#define BLOCK       256

typedef __attribute__((ext_vector_type(4))) float f4;

#if defined(__HIP_DEVICE_COMPILE__) && __has_builtin(__builtin_amdgcn_global_load_async_to_lds_b128)
#define HAVE_ASYNC_LDS 1
// Builtin signature (from hipcc diagnostic): pointee is a 16B int vector,
// first arg in address_space(1) (global), second in address_space(3) (LDS).
typedef int i4v __attribute__((vector_size(16)));
typedef __attribute__((address_space(1))) i4v as1_i4;
typedef __attribute__((address_space(3))) i4v as3_i4;
#else
#define HAVE_ASYNC_LDS 0
#endif

__global__ __launch_bounds__(BLOCK) void
PositionBias_26637387170463_kernel(const float* __restrict__ x,
                                   const int*   __restrict__ crop_x,
                                   const int*   __restrict__ crop_y,
                                   const float* __restrict__ bias,
                                   float*       __restrict__ out)
{
    // 32 KB bias table staged in LDS (fits easily in 320 KB/WGP).
    __shared__ __align__(16) float tbl[BIAS_ROWS * NUM_BINS];

    const unsigned t = threadIdx.x;

    // ---- Stage bias table: 8192 floats = 8 iters x 256 threads x float4 ----
#pragma unroll
    for (int i = 0; i < (BIAS_ROWS * NUM_BINS) / (BLOCK * 4); ++i) {
        const unsigned el = (unsigned)i * (BLOCK * 4) + t * 4u;
#if HAVE_ASYNC_LDS
        // gfx1250 async copy: global -> LDS, 16B per lane, tracked by ASYNCcnt
        // (no VGPR round trip).
        as1_i4* g = (as1_i4*)(uintptr_t)(bias + el);
        as3_i4* l = (as3_i4*)(uint32_t)(uintptr_t)(&tbl[el]);
        __builtin_amdgcn_global_load_async_to_lds_b128(g, l, /*offset=*/0, /*cpol=*/0);
#else
        *(f4*)&tbl[el] = *(const f4*)&bias[el];
#endif
    }

#if HAVE_ASYNC_LDS
#if __has_builtin(__builtin_amdgcn_s_wait_asynccnt)
    __builtin_amdgcn_s_wait_asynccnt(0);
#else
    asm volatile("s_wait_asynccnt 0" ::: "memory");
#endif
#endif
    __syncthreads();

    // ---- Main streaming loop: each block owns ROWS_PER_BLOCK (b,j) rows ----
    const int row0 = (int)blockIdx.x * ROWS_PER_BLOCK;      // row = b*256 + j
    const int bidx = row0 >> 8;                             // 256 rows per batch
    const int sd   = crop_x[bidx * 2] - crop_y[bidx * 2];   // scalar (SGPR) loads

    for (int r = 0; r < ROWS_PER_BLOCK; ++r) {
        const int    row  = row0 + r;
        const int    j    = row & (DIM - 1);
        const int    c    = sd - j;                         // diag = c + m
        const size_t base = (size_t)row * (DIM * NUM_BINS); // floats

#pragma unroll 4
        for (int i = 0; i < (DIM * NUM_BINS) / (BLOCK * 4); ++i) {  // 16 iters
            const int flat = i * BLOCK + (int)t;            // float4 index in row
            const int m    = flat >> 4;                     // 16 float4s per pixel
            const int q    = flat & 15;                     // which float4 of the 64 bins

            int d = c + m;
            d = (d < 0) ? -d : d;
            const int idx = (d < BIAS_ROWS - 1) ? d : (BIAS_ROWS - 1);

            // ds_load_b128 from the staged table (row = 256B, 16B-aligned chunks)
            const f4 bv = *(const f4*)&tbl[idx * NUM_BINS + q * 4];

            const size_t off = base + (size_t)flat * 4;
            // Streamed-once data: non-temporal so the bias table stays hot in cache.
            const f4 xv = __builtin_nontemporal_load((const f4*)(x + off));
            const f4 ov = xv + bv;
            __builtin_nontemporal_store(ov, (f4*)(out + off));
        }
    }
}

extern "C" void kernel_launch(void* const* d_in, const int* in_sizes, int n_in,
                              void* d_out, int out_size, void* d_ws, size_t ws_size,
                              hipStream_t stream) {
    (void)in_sizes; (void)n_in; (void)out_size; (void)d_ws; (void)ws_size;

    const float* x      = (const float*)d_in[0];   // [8,256,256,64] f32
    const int*   crop_x = (const int*)  d_in[1];   // [8,2] i32
    const int*   crop_y = (const int*)  d_in[2];   // [8,2] i32
    const float* bias   = (const float*)d_in[3];   // [128,64] f32
    float*       out    = (float*)d_out;           // [8,256,256,64] f32

    const int total_rows = 8 * DIM;                          // 2048
    dim3 grid(total_rows / ROWS_PER_BLOCK);                  // 512 blocks
    dim3 block(BLOCK);                                       // 8 waves (wave32)
    PositionBias_26637387170463_kernel<<<grid, block, 0, stream>>>(
        x, crop_x, crop_y, bias, out);
}